// Ranger_Averager_87978110091478
// MI455X (gfx1250) — compile-verified
//
#include <hip/hip_runtime.h>
#include <cstdint>
#include <cstddef>

// Ranger_Averager: out = fault ? truncated 3x3 box mean : x, fault = (x<-2.5)|(x>2.5)
// Tensor: [N=16, C=64, H=256, W=256] fp32. Pure HBM-bandwidth stencil (~23us floor
// @ 23.3 TB/s, AI ~ 1.6 FLOP/B -> far below any WMMA/VALU ceiling). MI455X-rewarded
// path: CDNA5 async global->LDS b128 copies (ASYNCcnt) for staging, LDS reuse for
// the 9-point window, non-temporal stores so the 256MB output stream doesn't
// thrash the 192MB L2.

#define HH 256
#define WW 256
#define STRIP 32
#define LROWS (STRIP + 2)   // 34 rows staged (strip + 1 halo row each side)

__device__ __forceinline__ void stage_chunk(const float* __restrict__ pin,
                                            float* __restrict__ tile,
                                            int r, int seg, int a /*absolute row*/) {
  float* ldsp = &tile[r * WW + seg * 4];
  if ((unsigned)a < (unsigned)HH) {
    // CDNA5 async copy, GVS addressing: SGPR64 plane base + VGPR32 byte offset.
    const uint32_t lds_off = (uint32_t)(uintptr_t)ldsp;          // low 32 bits = LDS offset
    const uint32_t voff    = (uint32_t)(a * WW + seg * 4) * 4u;  // byte offset in plane
    asm volatile("global_load_async_to_lds_b128 %0, %1, %2"
                 :: "v"(lds_off), "v"(voff), "s"(pin)
                 : "memory");
  } else {
    // Out-of-plane halo row: zero-fill (contributes 0 to truncated window sum,
    // the valid-cell count is handled analytically in the compute phase).
    *(float4*)ldsp = make_float4(0.f, 0.f, 0.f, 0.f);
  }
}

__global__ __launch_bounds__(256)
void ranger_box3_kernel(const float* __restrict__ in, float* __restrict__ out) {
  __shared__ float tile[LROWS * WW];     // 34 * 256 * 4B = 34 KB -> ~9 WGs/WGP

  const int tid   = threadIdx.x;         // 0..255
  const int strip = blockIdx.x;          // 0..7
  const int plane = blockIdx.y;          // 0..N*C-1
  const int y0    = strip * STRIP;       // first output row of strip
  const int base  = y0 - 1;              // absolute row held in LDS row 0 (may be -1)

  const float* __restrict__ pin  = in  + (size_t)plane * (HH * WW);
  float*       __restrict__ pout = out + (size_t)plane * (HH * WW);

  // ---------------- Stage tile: async global->LDS (16B per lane) ----------------
  // 34 rows * 64 segments = 2176 chunks. Uniform main loop: 8 iterations cover
  // chunks 0..2047 (rows 0..31); uniform tail: threads 0..127 cover 2048..2175
  // (rows 32..33). No per-lane trip-count divergence.
  const int seg = tid & 63;              // 16B segment within a row (0..63)
#pragma unroll
  for (int k = 0; k < 8; ++k) {
    const int r = k * 4 + (tid >> 6);    // LDS row 0..31
    stage_chunk(pin, tile, r, seg, base + r);
  }
  if (tid < 128) {
    const int r = 32 + (tid >> 6);       // LDS rows 32..33
    stage_chunk(pin, tile, r, seg, base + r);
  }
  // Drain this wave's async copies; barrier publishes all waves' data + ds stores.
  asm volatile("s_wait_asynccnt 0" ::: "memory");
  __syncthreads();

  // ---------------- Separable rolling 3x3 from LDS ----------------
  const int x = tid;                                      // this thread owns column x
  const float cnt_x = 1.f + (x > 0 ? 1.f : 0.f) + (x < WW - 1 ? 1.f : 0.f);

  // Horizontal 3-tap truncated sum of LDS row r at column x; also return center.
  auto hrow = [&](int r, float& center) -> float {
    const float* row = &tile[r * WW];
    const float c  = row[x];
    const float l  = (x > 0)      ? row[x - 1] : 0.f;
    const float rr = (x < WW - 1) ? row[x + 1] : 0.f;
    center = c;
    return l + c + rr;
  };

  float c_mid, c_top, dummy;
  float hm = hrow(0, dummy);        // row above first output row (zeroed if OOB)
  float hc = hrow(1, c_mid);        // first output row
  float hp = hrow(2, c_top);        // row below

#pragma unroll 4
  for (int yl = 0; yl < STRIP; ++yl) {
    const int ya = y0 + yl;
    const float cnt_y = 1.f + (ya > 0 ? 1.f : 0.f) + (ya < HH - 1 ? 1.f : 0.f);
    const float mean  = (hm + hc + hp) / (cnt_x * cnt_y);
    const float v     = c_mid;                         // original value at (ya, x)
    const float res   = (v < -2.5f || v > 2.5f) ? mean : v;
    __builtin_nontemporal_store(res, &pout[ya * WW + x]);

    // roll the vertical window down one row
    hm = hc; hc = hp; c_mid = c_top;
    if (yl + 1 < STRIP) hp = hrow(yl + 3, c_top);
  }
}

extern "C" void kernel_launch(void* const* d_in, const int* in_sizes, int n_in,
                              void* d_out, int out_size, void* d_ws, size_t ws_size,
                              hipStream_t stream) {
  (void)n_in; (void)out_size; (void)d_ws; (void)ws_size;
  const float* x = (const float*)d_in[0];
  float* out = (float*)d_out;
  const int planes = in_sizes[0] / (HH * WW);          // N*C = 1024
  dim3 grid(HH / STRIP, planes);                        // (8, 1024)
  ranger_box3_kernel<<<grid, 256, 0, stream>>>(x, out);
}